// CoreModel_88330297409891
// MI455X (gfx1250) — compile-verified
//
#include <hip/hip_runtime.h>
#include <cstddef>

typedef __attribute__((ext_vector_type(2))) float v2f;
typedef __attribute__((ext_vector_type(8))) float v8f;

// ---------------- workspace layout (floats) ----------------
constexpr size_t OXQ   = 0;                  // 16x1024  [s_prev|e_t]
constexpr size_t OHQ   = OXQ   + 16*1024;    // 16x1024  q hidden
constexpr size_t OQ    = OHQ   + 16*1024;    // 16x128   q
constexpr size_t OCT   = OQ    + 16*128;     // 16x256   c_t
constexpr size_t OXZ   = OCT   + 16*256;     // 16x768   [e_t|c_t]
constexpr size_t OXG   = OXZ   + 16*768;     // 16x1280  [s_prev|e_t|c_t]
constexpr size_t OXB   = OXG   + 16*1280;    // 16x512   [c_t|z_t]
constexpr size_t OGT   = OXB   + 16*512;     // 16x512   g_t
constexpr size_t OMODU = OGT   + 16*512;     // 16x256
constexpr size_t OBV   = OMODU + 16*256;     // 16x256
constexpr size_t OWT   = OBV   + 16*256;     // 16x256   w_t
constexpr size_t OXP   = OWT   + 16*256;     // 16x512   [p_prev|c_t]
constexpr size_t OHP   = OXP   + 16*512;     // 16x512
constexpr size_t OPT   = OHP   + 16*512;     // 16x256   p_t
constexpr size_t OWI   = OPT   + 16*256;     // 16x1024  [s_t|e_t]
constexpr size_t OGW   = OWI   + 16*1024;    // 32       gw[3][8]
constexpr size_t OVW   = OGW   + 32;         // 3x16x256
constexpr size_t OKW   = OVW   + 3*16*256;   // 3x16x128
constexpr size_t OCTXS = OKW   + 3*16*128;   // 3x16x256
constexpr size_t ACT_END = OCTXS + 3*16*256;
constexpr size_t OP0   = ACT_END;            // 8x64x258 attention partials L0
constexpr size_t OP1   = OP0 + (size_t)8*64*258;
constexpr size_t OP2   = OP1 + (size_t)8*16*258;
constexpr size_t WS_FLOATS = OP2 + (size_t)8*4*258;

#define CHUNK 512

// ---------------- 16-row GEMM via V_WMMA_F32_16X16X4_F32 ----------------
// Y[r, n0+n] = act( sum_k X[r,k]*W[k,n] + bias[n] ) (+ residual), rows 0..7 only.
__global__ void gemm16_wmma(const float* __restrict__ X, int Kdim,
                            const float* __restrict__ W, int N,
                            const float* __restrict__ bias,
                            const float* __restrict__ residual, int ldRes,
                            float* __restrict__ Y, int ldY, int act)
{
    int tile = blockIdx.x * (blockDim.x >> 5) + (threadIdx.x >> 5);
    if (tile * 16 >= N) return;
    int n0   = tile * 16;
    int lane = threadIdx.x & 31;
    int m    = lane & 15;
    int half = lane >> 4;
    float res[8];
#if __has_builtin(__builtin_amdgcn_wmma_f32_16x16x4_f32)
    v8f acc = {0.f,0.f,0.f,0.f,0.f,0.f,0.f,0.f};
    for (int k0 = 0; k0 < Kdim; k0 += 4) {
        int ka = k0 + half * 2;
        v2f a, b;
        a.x = X[(size_t)m * Kdim + ka];
        a.y = X[(size_t)m * Kdim + ka + 1];
        b.x = W[(size_t)ka * N + n0 + m];
        b.y = W[(size_t)(ka + 1) * N + n0 + m];
        acc = __builtin_amdgcn_wmma_f32_16x16x4_f32(false, a, false, b,
                                                    (short)0, acc, false, false);
    }
    #pragma unroll
    for (int r = 0; r < 8; ++r) res[r] = acc[r];
#else
    // scalar fallback (half==0 lanes produce rows 0..7 for column n0+m)
    #pragma unroll
    for (int r = 0; r < 8; ++r) res[r] = 0.f;
    for (int k = 0; k < Kdim; ++k) {
        float w = W[(size_t)k * N + n0 + m];
        #pragma unroll
        for (int r = 0; r < 8; ++r) res[r] += X[(size_t)r * Kdim + k] * w;
    }
#endif
    if (half == 0) {
        int n = n0 + m;
        float bv = bias ? bias[n] : 0.f;
        #pragma unroll
        for (int r = 0; r < 8; ++r) {
            float y = res[r] + bv;
            if (act == 1)      y = 1.f / (1.f + __expf(-y));
            else if (act == 2) y = tanhf(y);
            if (residual)      y += residual[(size_t)r * ldRes + n];
            Y[(size_t)r * ldY + n] = y;
        }
    }
}

// ---------------- LayerNorm + ReLU (in place, 8 rows) ----------------
__global__ void ln_relu(float* __restrict__ Xbuf, int width,
                        const float* __restrict__ g, const float* __restrict__ beta)
{
    int b = blockIdx.x;
    float* row = Xbuf + (size_t)b * width;
    __shared__ float red[256];
    int tid = threadIdx.x;
    float s = 0.f;
    for (int i = tid; i < width; i += 256) s += row[i];
    red[tid] = s; __syncthreads();
    for (int off = 128; off > 0; off >>= 1) { if (tid < off) red[tid] += red[tid+off]; __syncthreads(); }
    float mean = red[0] / (float)width; __syncthreads();
    float v = 0.f;
    for (int i = tid; i < width; i += 256) { float d = row[i] - mean; v += d*d; }
    red[tid] = v; __syncthreads();
    for (int off = 128; off > 0; off >>= 1) { if (tid < off) red[tid] += red[tid+off]; __syncthreads(); }
    float rstd = rsqrtf(red[0] / (float)width + 1e-6f); __syncthreads();
    for (int i = tid; i < width; i += 256) {
        float y = (row[i] - mean) * rstd * g[i] + beta[i];
        row[i] = y > 0.f ? y : 0.f;
    }
}

// ---------------- static concat builder ----------------
__global__ void build_static(const float* __restrict__ s_prev, const float* __restrict__ e_t,
                             const float* __restrict__ p_prev,
                             float* __restrict__ xq, float* __restrict__ xz,
                             float* __restrict__ xg, float* __restrict__ xp,
                             float* __restrict__ wi)
{
    int b = blockIdx.x, tid = threadIdx.x;
    for (int i = tid; i < 512; i += 256) {
        float sv = s_prev[(size_t)b*512 + i];
        float ev = e_t  [(size_t)b*512 + i];
        xq[(size_t)b*1024 + i]        = sv;
        xq[(size_t)b*1024 + 512 + i]  = ev;
        xz[(size_t)b*768  + i]        = ev;
        xg[(size_t)b*1280 + i]        = sv;
        xg[(size_t)b*1280 + 512 + i]  = ev;
        wi[(size_t)b*1024 + 512 + i]  = ev;
    }
    xp[(size_t)b*512 + tid] = p_prev[(size_t)b*256 + tid];
}

// ---------------- attention: per-chunk online-softmax partials ----------------
__global__ void attn_partial(const float* __restrict__ Kmat, const float* __restrict__ Mmat,
                             const float* __restrict__ q, float* __restrict__ part,
                             int N, int nchunks)
{
    int b = blockIdx.y, chunk = blockIdx.x;
    int n0 = chunk * CHUNK;
    const float* Kb = Kmat + (size_t)b * N * 128;
    const float* Mb = Mmat + (size_t)b * N * 256;
    __shared__ float qs[128];
    __shared__ float sc[CHUNK];
    __shared__ float red[16];
    int tid = threadIdx.x, wave = tid >> 5, lane = tid & 31;
    if (tid < 128) qs[tid] = q[(size_t)b * 128 + tid];
    __syncthreads();
    const float scale = 0.08838834764831845f; // 1/sqrt(128)
    float lmax = -3.402823466e38f;
    for (int row = wave; row < CHUNK; row += 8) {
        const float4 kv = reinterpret_cast<const float4*>(Kb + (size_t)(n0 + row) * 128)[lane];
        float p = kv.x * qs[lane*4] + kv.y * qs[lane*4+1]
                + kv.z * qs[lane*4+2] + kv.w * qs[lane*4+3];
        #pragma unroll
        for (int off = 16; off > 0; off >>= 1) p += __shfl_xor(p, off, 32);
        float sres = p * scale;
        if (lane == 0) sc[row] = sres;
        lmax = fmaxf(lmax, sres);
    }
    if (lane == 0) red[wave] = lmax;
    __syncthreads();
    if (tid == 0) {
        float m = red[0];
        for (int w = 1; w < 8; ++w) m = fmaxf(m, red[w]);
        red[8] = m;
    }
    __syncthreads();
    float mloc = red[8];
    float zp = 0.f;
    for (int i = tid; i < CHUNK; i += 256) {
        float w = __expf(sc[i] - mloc);
        sc[i] = w;
        zp += w;
    }
    #pragma unroll
    for (int off = 16; off > 0; off >>= 1) zp += __shfl_xor(zp, off, 32);
    if (lane == 0) red[wave] = zp;
    __syncthreads();
    if (tid == 0) {
        float z = 0.f;
        for (int w = 0; w < 8; ++w) z += red[w];
        red[9] = z;
    }
    __syncthreads();
    // column-parallel weighted sum of M rows (coalesced)
    float acc = 0.f;
    const float* Mrow = Mb + (size_t)n0 * 256 + tid;
    #pragma unroll 8
    for (int r = 0; r < CHUNK; ++r) acc += sc[r] * Mrow[(size_t)r * 256];
    float* base = part + ((size_t)b * nchunks + chunk) * 258;
    if (tid == 0) { base[0] = mloc; base[1] = red[9]; }
    base[2 + tid] = acc;
}

// ---------------- combine partials + project ctx ----------------
__global__ void attn_reduce(const float* __restrict__ P0, const float* __restrict__ P1,
                            const float* __restrict__ P2,
                            const float* __restrict__ Wp0, const float* __restrict__ Wp1,
                            const float* __restrict__ Wp2,
                            float* __restrict__ ctxs, int nb0, int nb1, int nb2)
{
    int b = blockIdx.x, l = blockIdx.y, tid = threadIdx.x;
    const float* P  = (l==0) ? P0  : (l==1) ? P1  : P2;
    const float* Wp = (l==0) ? Wp0 : (l==1) ? Wp1 : Wp2;
    int nb          = (l==0) ? nb0 : (l==1) ? nb1 : nb2;
    const float* base = P + (size_t)b * nb * 258;
    __shared__ float red[256];
    __shared__ float ctxS[256];
    __shared__ float gmS, zS;
    float mx = -3.402823466e38f;
    for (int i = tid; i < nb; i += 256) mx = fmaxf(mx, base[(size_t)i*258]);
    red[tid] = mx; __syncthreads();
    for (int off = 128; off > 0; off >>= 1) { if (tid < off) red[tid] = fmaxf(red[tid], red[tid+off]); __syncthreads(); }
    if (tid == 0) gmS = red[0];
    __syncthreads();
    float gm = gmS;
    float z = 0.f;
    for (int i = tid; i < nb; i += 256) z += base[(size_t)i*258+1] * __expf(base[(size_t)i*258] - gm);
    red[tid] = z; __syncthreads();
    for (int off = 128; off > 0; off >>= 1) { if (tid < off) red[tid] += red[tid+off]; __syncthreads(); }
    if (tid == 0) zS = red[0];
    __syncthreads();
    float Z = zS;
    float acc = 0.f;
    for (int i = 0; i < nb; ++i) acc += base[(size_t)i*258 + 2 + tid] * __expf(base[(size_t)i*258] - gm);
    ctxS[tid] = acc / Z;
    __syncthreads();
    float s = 0.f;
    for (int k = 0; k < 256; ++k) s += ctxS[k] * Wp[(size_t)k*256 + tid];
    ctxs[((size_t)l*16 + b)*256 + tid] = s;
}

// ---------------- mix softmax + c_t scatter ----------------
__global__ void mix_ct(const float* __restrict__ ctxs, const float* __restrict__ Wmix,
                       const float* __restrict__ bmix,
                       float* __restrict__ ct, float* __restrict__ xz, float* __restrict__ xg,
                       float* __restrict__ xB, float* __restrict__ xp)
{
    int b = blockIdx.x, tid = threadIdx.x;
    __shared__ float r0[256], r1[256], r2[256];
    __shared__ float mixS[3];
    float p0 = 0.f, p1 = 0.f, p2 = 0.f;
    for (int k = tid; k < 768; k += 256) {
        float v = ctxs[((size_t)(k>>8)*16 + b)*256 + (k & 255)];
        p0 += v * Wmix[k*3+0]; p1 += v * Wmix[k*3+1]; p2 += v * Wmix[k*3+2];
    }
    r0[tid]=p0; r1[tid]=p1; r2[tid]=p2; __syncthreads();
    for (int off = 128; off > 0; off >>= 1) {
        if (tid < off) { r0[tid]+=r0[tid+off]; r1[tid]+=r1[tid+off]; r2[tid]+=r2[tid+off]; }
        __syncthreads();
    }
    if (tid == 0) {
        float l0 = r0[0]+bmix[0], l1 = r1[0]+bmix[1], l2 = r2[0]+bmix[2];
        float m = fmaxf(l0, fmaxf(l1, l2));
        float e0 = __expf(l0-m), e1 = __expf(l1-m), e2 = __expf(l2-m);
        float inv = 1.f / (e0+e1+e2);
        mixS[0]=e0*inv; mixS[1]=e1*inv; mixS[2]=e2*inv;
    }
    __syncthreads();
    float c = mixS[0]*ctxs[((size_t)0*16+b)*256+tid]
            + mixS[1]*ctxs[((size_t)1*16+b)*256+tid]
            + mixS[2]*ctxs[((size_t)2*16+b)*256+tid];
    ct[(size_t)b*256 + tid]        = c;
    xz[(size_t)b*768 + 512 + tid]  = c;
    xg[(size_t)b*1280 + 1024 + tid]= c;
    xB[(size_t)b*512 + tid]        = c;
    xp[(size_t)b*512 + 256 + tid]  = c;
}

// ---------------- wave subsystem: w_t = modu*(A_base w_prev) + Bv ----------------
__global__ void wave_k(const float* __restrict__ w_prev, const float* __restrict__ A_diag,
                       const float* __restrict__ A_U, const float* __restrict__ A_V,
                       const float* __restrict__ modu, const float* __restrict__ Bv,
                       float* __restrict__ w_t)
{
    int b = blockIdx.x, tid = threadIdx.x;
    __shared__ float wp[256], tS[64];
    wp[tid] = w_prev[(size_t)b*256 + tid];
    __syncthreads();
    if (tid < 64) {
        float s = 0.f;
        for (int i = 0; i < 256; ++i) s += A_V[(size_t)i*64 + tid] * wp[i];
        tS[tid] = s;
    }
    __syncthreads();
    float u = 0.f;
    for (int j = 0; j < 64; ++j) u += A_U[(size_t)tid*64 + j] * tS[j];
    float ad = tanhf(A_diag[tid]) * 0.9f;
    w_t[(size_t)b*256 + tid] = modu[(size_t)b*256 + tid] * (ad*wp[tid] + u) + Bv[(size_t)b*256 + tid];
}

// ---------------- gated state s_t + wi concat ----------------
__global__ void st_wi(const float* __restrict__ s_prev, const float* __restrict__ g_t,
                      const float* __restrict__ p_t, const float* __restrict__ w_t,
                      const float* __restrict__ U_p, const float* __restrict__ U_w,
                      float* __restrict__ wi, float* __restrict__ out_s)
{
    int b = blockIdx.x, i = threadIdx.x; // blockDim 512
    __shared__ float pS[256], wS[256];
    if (i < 256) { pS[i] = p_t[(size_t)b*256+i]; wS[i] = w_t[(size_t)b*256+i]; }
    __syncthreads();
    float sp = 0.f, sw = 0.f;
    for (int j = 0; j < 256; ++j) {
        sp += U_p[(size_t)i*256 + j] * pS[j];
        sw += U_w[(size_t)i*256 + j] * wS[j];
    }
    float g = g_t[(size_t)b*512 + i];
    float s = s_prev[(size_t)b*512 + i] + g*sp + (1.f - g)*sw;
    wi[(size_t)b*1024 + i] = s;
    out_s[(size_t)b*512 + i] = s;
}

// ---------------- write gates gw[l][b] ----------------
__global__ void gw_k(const float* __restrict__ wi, const float* __restrict__ r_t,
                     const float* __restrict__ Wg0, const float* __restrict__ Wg1,
                     const float* __restrict__ Wg2,
                     const float* __restrict__ bg0, const float* __restrict__ bg1,
                     const float* __restrict__ bg2,
                     float* __restrict__ gw)
{
    int l = blockIdx.x, b = blockIdx.y, tid = threadIdx.x; // blockDim 128
    const float* Wg = (l==0) ? Wg0 : (l==1) ? Wg1 : Wg2;
    const float* bg = (l==0) ? bg0 : (l==1) ? bg1 : bg2;
    __shared__ float red[128];
    float s = 0.f;
    for (int k = tid; k < 1024; k += 128) s += wi[(size_t)b*1024 + k] * Wg[k];
    red[tid] = s; __syncthreads();
    for (int off = 64; off > 0; off >>= 1) { if (tid < off) red[tid] += red[tid+off]; __syncthreads(); }
    if (tid == 0) {
        float tot = red[0] + r_t[b]*Wg[1024] + bg[0];
        gw[l*8 + b] = 1.f / (1.f + __expf(-tot));
    }
}

// ---------------- streaming decay (the 1 GB of traffic) ----------------
__global__ void scale_k(const float4* __restrict__ in, float4* __restrict__ out,
                        float keep, int n4)
{
    int i = blockIdx.x * 256 + threadIdx.x;
    if (i < n4) {
        float4 v = in[i];
        v.x *= keep; v.y *= keep; v.z *= keep; v.w *= keep;
        out[i] = v;
    }
}

// ---------------- row-0 gated write ----------------
__global__ void row0_k(const float* __restrict__ Min, const float* __restrict__ Kin,
                       float* __restrict__ Mout, float* __restrict__ Kout,
                       const float* __restrict__ vw, const float* __restrict__ kw,
                       const float* __restrict__ gw, float keep, int N)
{
    int b = blockIdx.x, d = threadIdx.x; // 256
    float g = gw[b];
    size_t mo = (size_t)b * N * 256 + d;
    Mout[mo] = g * vw[(size_t)b*256 + d] + (1.f - g) * (Min[mo] * keep);
    if (d < 128) {
        size_t ko = (size_t)b * N * 128 + d;
        Kout[ko] = g * kw[(size_t)b*128 + d] + (1.f - g) * (Kin[ko] * keep);
    }
}

extern "C" void kernel_launch(void* const* d_in, const int* in_sizes, int n_in,
                              void* d_out, int out_size, void* d_ws, size_t ws_size,
                              hipStream_t stream)
{
    // inputs in setup_inputs() insertion order
    const float* M0 = (const float*)d_in[0];
    const float* K0 = (const float*)d_in[1];
    const float* M1 = (const float*)d_in[2];
    const float* K1 = (const float*)d_in[3];
    const float* M2 = (const float*)d_in[4];
    const float* K2 = (const float*)d_in[5];
    const float* s_prev = (const float*)d_in[6];
    const float* w_prev = (const float*)d_in[7];
    const float* p_prev = (const float*)d_in[8];
    const float* e_t = (const float*)d_in[9];
    const float* r_t = (const float*)d_in[10];
    // params (insertion order)
    const float* Wq1 = (const float*)d_in[11];
    const float* bq1 = (const float*)d_in[12];
    const float* gq  = (const float*)d_in[13];
    const float* betq= (const float*)d_in[14];
    const float* Wq2 = (const float*)d_in[15];
    const float* bq2 = (const float*)d_in[16];
    const float* Wproj0 = (const float*)d_in[17];
    const float* Wproj1 = (const float*)d_in[18];
    const float* Wproj2 = (const float*)d_in[19];
    const float* Wmix = (const float*)d_in[20];
    const float* bmix = (const float*)d_in[21];
    const float* Wz = (const float*)d_in[22];
    const float* bz = (const float*)d_in[23];
    const float* Wg = (const float*)d_in[24];
    const float* bg = (const float*)d_in[25];
    const float* U_p = (const float*)d_in[26];
    const float* U_w = (const float*)d_in[27];
    const float* A_diag = (const float*)d_in[28];
    const float* A_U = (const float*)d_in[29];
    const float* A_V = (const float*)d_in[30];
    const float* Wmod = (const float*)d_in[31];
    const float* bmod = (const float*)d_in[32];
    const float* WB = (const float*)d_in[33];
    const float* bB = (const float*)d_in[34];
    const float* Wp1 = (const float*)d_in[35];
    const float* bp1 = (const float*)d_in[36];
    const float* gp  = (const float*)d_in[37];
    const float* betp= (const float*)d_in[38];
    const float* Wp2 = (const float*)d_in[39];
    const float* bp2 = (const float*)d_in[40];
    const float* Wwg0 = (const float*)d_in[41];
    const float* Wwg1 = (const float*)d_in[42];
    const float* Wwg2 = (const float*)d_in[43];
    const float* bwg0 = (const float*)d_in[44];
    const float* bwg1 = (const float*)d_in[45];
    const float* bwg2 = (const float*)d_in[46];
    const float* Wwv0 = (const float*)d_in[47];
    const float* Wwv1 = (const float*)d_in[48];
    const float* Wwv2 = (const float*)d_in[49];
    const float* bwv0 = (const float*)d_in[50];
    const float* bwv1 = (const float*)d_in[51];
    const float* bwv2 = (const float*)d_in[52];
    const float* Wwk0 = (const float*)d_in[53];
    const float* Wwk1 = (const float*)d_in[54];
    const float* Wwk2 = (const float*)d_in[55];
    const float* bwk0 = (const float*)d_in[56];
    const float* bwk1 = (const float*)d_in[57];
    const float* bwk2 = (const float*)d_in[58];

    float* ws = (float*)d_ws;
    float* out = (float*)d_out;
    // output offsets (return order: s_t, M0,M1,M2, K0,K1,K2)
    const size_t oM0 = 4096;
    const size_t oM1 = oM0 + (size_t)8*32768*256;
    const size_t oM2 = oM1 + (size_t)8*8192*256;
    const size_t oK0 = oM2 + (size_t)8*2048*256;
    const size_t oK1 = oK0 + (size_t)8*32768*128;
    const size_t oK2 = oK1 + (size_t)8*8192*128;

    // zero activation scratch so padded rows 8..15 of every GEMM input are 0
    hipMemsetAsync(d_ws, 0, ACT_END * sizeof(float), stream);

    auto gemm = [&](const float* X, int Kd, const float* W, int N, const float* bias,
                    const float* res, int ldRes, float* Y, int ldY, int act) {
        int grid = (N / 16 + 7) / 8;
        gemm16_wmma<<<dim3(grid), dim3(256), 0, stream>>>(X, Kd, W, N, bias, res, ldRes, Y, ldY, act);
    };

    build_static<<<8, 256, 0, stream>>>(s_prev, e_t, p_prev,
                                        ws+OXQ, ws+OXZ, ws+OXG, ws+OXP, ws+OWI);

    // ---- QueryNetwork ----
    gemm(ws+OXQ, 1024, Wq1, 1024, bq1, nullptr, 0, ws+OHQ, 1024, 0);
    ln_relu<<<8, 256, 0, stream>>>(ws+OHQ, 1024, gq, betq);
    gemm(ws+OHQ, 1024, Wq2, 128, bq2, nullptr, 0, ws+OQ, 128, 0);

    // ---- CMSRead attention (single pass over K and M) ----
    attn_partial<<<dim3(64, 8), 256, 0, stream>>>(K0, M0, ws+OQ, ws+OP0, 32768, 64);
    attn_partial<<<dim3(16, 8), 256, 0, stream>>>(K1, M1, ws+OQ, ws+OP1, 8192, 16);
    attn_partial<<<dim3(4, 8),  256, 0, stream>>>(K2, M2, ws+OQ, ws+OP2, 2048, 4);
    attn_reduce<<<dim3(8, 3), 256, 0, stream>>>(ws+OP0, ws+OP1, ws+OP2,
                                                Wproj0, Wproj1, Wproj2,
                                                ws+OCTXS, 64, 16, 4);
    mix_ct<<<8, 256, 0, stream>>>(ws+OCTXS, Wmix, bmix,
                                  ws+OCT, ws+OXZ, ws+OXG, ws+OXB, ws+OXP);

    // ---- HOPECore ----
    gemm(ws+OXZ, 768,  Wz,   256, bz,   nullptr, 0, ws+OXB+256, 512, 0); // z_t -> xB[:,256:]
    gemm(ws+OXG, 1280, Wg,   512, bg,   nullptr, 0, ws+OGT, 512, 1);     // sigmoid
    gemm(ws+OCT, 256,  Wmod, 256, bmod, nullptr, 0, ws+OMODU, 256, 2);   // tanh
    gemm(ws+OXB, 512,  WB,   256, bB,   nullptr, 0, ws+OBV, 256, 0);
    wave_k<<<8, 256, 0, stream>>>(w_prev, A_diag, A_U, A_V, ws+OMODU, ws+OBV, ws+OWT);
    gemm(ws+OXP, 512,  Wp1,  512, bp1,  nullptr, 0, ws+OHP, 512, 0);
    ln_relu<<<8, 256, 0, stream>>>(ws+OHP, 512, gp, betp);
    gemm(ws+OHP, 512,  Wp2,  256, bp2,  p_prev, 256, ws+OPT, 256, 0);    // + residual p_prev
    st_wi<<<8, 512, 0, stream>>>(s_prev, ws+OGT, ws+OPT, ws+OWT, U_p, U_w,
                                 ws+OWI, out /* s_t */);

    // ---- CMSWrite small projections ----
    gemm(ws+OWI, 1024, Wwv0, 256, bwv0, nullptr, 0, ws+OVW + 0*16*256, 256, 0);
    gemm(ws+OWI, 1024, Wwv1, 256, bwv1, nullptr, 0, ws+OVW + 1*16*256, 256, 0);
    gemm(ws+OWI, 1024, Wwv2, 256, bwv2, nullptr, 0, ws+OVW + 2*16*256, 256, 0);
    gemm(ws+OWI, 1024, Wwk0, 128, bwk0, nullptr, 0, ws+OKW + 0*16*128, 128, 0);
    gemm(ws+OWI, 1024, Wwk1, 128, bwk1, nullptr, 0, ws+OKW + 1*16*128, 128, 0);
    gemm(ws+OWI, 1024, Wwk2, 128, bwk2, nullptr, 0, ws+OKW + 2*16*128, 128, 0);
    gw_k<<<dim3(3, 8), 128, 0, stream>>>(ws+OWI, r_t, Wwg0, Wwg1, Wwg2,
                                         bwg0, bwg1, bwg2, ws+OGW);

    // ---- streaming decay + row-0 gated write ----
    const float kp0 = 0.999f, kp1 = 0.99f, kp2 = 0.9f;
    auto scale = [&](const float* in, float* o, float keep, size_t n) {
        int n4 = (int)(n / 4);
        scale_k<<<dim3((n4 + 255) / 256), dim3(256), 0, stream>>>(
            (const float4*)in, (float4*)o, keep, n4);
    };
    scale(M0, out + oM0, kp0, (size_t)8*32768*256);
    scale(K0, out + oK0, kp0, (size_t)8*32768*128);
    scale(M1, out + oM1, kp1, (size_t)8*8192*256);
    scale(K1, out + oK1, kp1, (size_t)8*8192*128);
    scale(M2, out + oM2, kp2, (size_t)8*2048*256);
    scale(K2, out + oK2, kp2, (size_t)8*2048*128);
    row0_k<<<8, 256, 0, stream>>>(M0, K0, out + oM0, out + oK0,
                                  ws+OVW + 0*16*256, ws+OKW + 0*16*128, ws+OGW + 0, kp0, 32768);
    row0_k<<<8, 256, 0, stream>>>(M1, K1, out + oM1, out + oK1,
                                  ws+OVW + 1*16*256, ws+OKW + 1*16*128, ws+OGW + 8, kp1, 8192);
    row0_k<<<8, 256, 0, stream>>>(M2, K2, out + oM2, out + oK2,
                                  ws+OVW + 2*16*256, ws+OKW + 2*16*128, ws+OGW + 16, kp2, 2048);
    (void)in_sizes; (void)n_in; (void)out_size; (void)ws_size;
}